// LogEigLayer_23433341567258
// MI455X (gfx1250) — compile-verified
//
#include <hip/hip_runtime.h>
#include <hip/hip_bf16.h>
#include <math.h>

// Batched 64x64 SPD matrix logarithm on gfx1250, GEMM-only
// (inverse scaling & squaring + coupled Newton-Schulz square roots),
// all matmuls on the f32 matrix pipe via V_WMMA_F32_16X16X4_F32.
// All working-set traffic is LDS (ds_*): buffers are selected by integer
// element offsets into a single __shared__ array so addrspace(3) inference
// succeeds (a dynamically indexed pointer array forces FLAT codegen).

namespace {

constexpr int MD    = 64;        // matrix dimension
constexpr int LDST  = 66;        // even padded stride: 8B-aligned K-pairs,
                                 // conflict-free C stores
constexpr int SZ    = MD * LDST; // floats per LDS matrix buffer
constexpr int NBUF  = 5;         // Y, Z, T, Y', Z'
constexpr int KSQRT = 6;         // #square roots; rescale factor 2^6 = 64

typedef __attribute__((ext_vector_type(2))) float v2f;
typedef __attribute__((ext_vector_type(8))) float v8f;

enum { EPI_COPY = 0, EPI_NS = 1, EPI_DIAG = 2, EPI_FINAL = 3 };

// C(16x16 tiles ti, tj0..tj0+NTJ-1) = A*B for 64x64 LDS matrices addressed
// as lds[aoff + ...] / lds[boff + ...]. K is chained through 16
// V_WMMA_F32_16X16X4_F32 in two chunks of 8: batch ds-loads into registers,
// then run the WMMAs back-to-back (one DScnt wait per chunk, not per WMMA).
// Operand layouts per CDNA5 ISA 7.12.2:
//   A 16x4 f32 : v[0] lanes0-15 K=0 / lanes16-31 K=2 ; v[1] K=1 / K=3
//   B  4x16 f32: v[0] lanes0-15 row K=0 / lanes16-31 row K=2 ; v[1] K=1/K=3
//   C 16x16 f32: v[r] lanes0-15 M=r / lanes16-31 M=r+8
template <int NTJ, int EPI>
__device__ __forceinline__ void gemm_tiles(float* lds, int aoff, int boff,
                                           int coff, float* __restrict__ gC,
                                           int ti, int tj0, int lane,
                                           float p0, float p1)
{
  const int half = lane >> 4;   // K sub-pair selector {0,1} vs {2,3}
  const int l    = lane & 15;
  v8f acc[NTJ] = {};
  const int abase = aoff + (ti * 16 + l) * LDST + half * 2; // 8B aligned
  const int bbase = boff + (half * 2) * LDST + l;
#pragma unroll
  for (int kc = 0; kc < 2; ++kc) {
    v2f a[8];
    v2f b[NTJ][8];
#pragma unroll
    for (int kk = 0; kk < 8; ++kk) {
      const int k4 = (kc * 8 + kk) * 4;
      a[kk] = *reinterpret_cast<const v2f*>(lds + abase + k4); // ds_load_b64
#pragma unroll
      for (int j = 0; j < NTJ; ++j) {
        const int bo = bbase + (tj0 + j) * 16 + k4 * LDST;
        b[j][kk].x = lds[bo];
        b[j][kk].y = lds[bo + LDST];
      }
    }
#pragma unroll
    for (int kk = 0; kk < 8; ++kk) {
#pragma unroll
      for (int j = 0; j < NTJ; ++j) {
        acc[j] = __builtin_amdgcn_wmma_f32_16x16x4_f32(
            false, a[kk], false, b[j][kk], (short)0, acc[j], false, false);
      }
    }
  }
  const int row0 = ti * 16 + half * 8;
#pragma unroll
  for (int j = 0; j < NTJ; ++j) {
    const int col = (tj0 + j) * 16 + l;
#pragma unroll
    for (int r = 0; r < 8; ++r) {
      const int row = row0 + r;
      float v = acc[j][r];
      if (EPI == EPI_NS)    v = ((row == col) ? 1.5f : 0.0f) - 0.5f * v;
      if (EPI == EPI_DIAG)  v = v + ((row == col) ? p0 : 0.0f);
      if (EPI == EPI_FINAL) v = p0 * v + ((row == col) ? p1 : 0.0f);
      if (EPI == EPI_FINAL) gC[row * MD + col] = v;        // global b32 store
      else                  lds[coff + row * LDST + col] = v; // ds_store_b32
    }
  }
}

} // namespace

__global__ __launch_bounds__(256) void logm64_wmma_kernel(
    const float* __restrict__ X, float* __restrict__ Out)
{
  __shared__ __align__(16) float lds[NBUF * SZ];   // 84,480 B (< 320 KB/WGP)
  __shared__ float s_fro;

  const int tid  = threadIdx.x;
  const int wv   = __builtin_amdgcn_readfirstlane(tid) >> 5; // scalar wave id
  const int lane = tid & 31;
  const size_t base = (size_t)blockIdx.x * (MD * MD);
  const float* Xb = X + base;
  float* Ob = Out + base;

  if (tid == 0) s_fro = 0.0f;

  // ---- load raw X into buffer 2 (16 B / thread / iter, coalesced b128) ----
  const int RAW = 2 * SZ;
#pragma unroll
  for (int i = 0; i < 4; ++i) {
    const int f4 = tid + 256 * i;           // 1024 float4 per matrix
    const int e  = f4 * 4;
    const int r  = e >> 6, c = e & 63;
    const float4 v = ((const float4*)Xb)[f4];
    lds[RAW + r * LDST + c + 0] = v.x;
    lds[RAW + r * LDST + c + 1] = v.y;
    lds[RAW + r * LDST + c + 2] = v.z;
    lds[RAW + r * LDST + c + 3] = v.w;
  }
  __syncthreads();

  // ---- symmetrize + Frobenius norm (spectral upper bound) ----
  float xs[16];
  float part = 0.0f;
#pragma unroll
  for (int i = 0; i < 16; ++i) {
    const int e = tid + 256 * i;
    const int r = e >> 6, c = e & 63;
    const float v = 0.5f * (lds[RAW + r * LDST + c] + lds[RAW + c * LDST + r]);
    xs[i] = v;
    part += v * v;
  }
  atomicAdd(&s_fro, part);   // ds_add_f32
  __syncthreads();
  const float s    = sqrtf(s_fro);
  const float invs = 1.0f / s;
  const float logs = logf(s);

  // ---- Y = Xs/s (eigenvalues in (0,1]) ; Z = I ----
#pragma unroll
  for (int i = 0; i < 16; ++i) {
    const int e = tid + 256 * i;
    const int r = e >> 6, c = e & 63;
    lds[0 * SZ + r * LDST + c] = xs[i] * invs;
    lds[1 * SZ + r * LDST + c] = (r == c) ? 1.0f : 0.0f;
  }
  __syncthreads();

  int yb = 0 * SZ, zb = 1 * SZ, y2 = 3 * SZ, z2 = 4 * SZ;
  const int tb = 2 * SZ;

  // ---- K chained square roots via coupled Newton-Schulz ----
  for (int step = 0; step < KSQRT; ++step) {
    const int nit = (step == 0) ? 16 : (step == 1) ? 11 : (step == 2) ? 9
                  : (step == 3) ? 8  : (step == 4) ? 7  : 6;
    if (step > 0) {
      // reseed Z = I for the next sqrt (Y holds the current iterate)
#pragma unroll
      for (int i = 0; i < 16; ++i) {
        const int e = tid + 256 * i;
        const int r = e >> 6, c = e & 63;
        lds[zb + r * LDST + c] = (r == c) ? 1.0f : 0.0f;
      }
      __syncthreads();
    }
    for (int it = 0; it < nit; ++it) {
      // T = 1.5 I - 0.5 * Z*Y   (8 waves x 2 tiles, NS epilogue fused)
      gemm_tiles<2, EPI_NS>(lds, zb, yb, tb, nullptr,
                            wv & 3, (wv >> 2) * 2, lane, 0.f, 0.f);
      __syncthreads();
      // Y' = Y*T (waves 0..3) ; Z' = T*Z (waves 4..7) — concurrent GEMMs
      if (wv < 4) {
        gemm_tiles<4, EPI_COPY>(lds, yb, tb, y2, nullptr,
                                wv, 0, lane, 0.f, 0.f);
      } else {
        gemm_tiles<4, EPI_COPY>(lds, tb, zb, z2, nullptr,
                                wv - 4, 0, lane, 0.f, 0.f);
      }
      __syncthreads();
      int t;
      t = yb; yb = y2; y2 = t;
      t = zb; zb = z2; z2 = t;
    }
  }

  // ---- E = A^(1/64) - I ; first Horner step fused elementwise:
  //      W = c8*E + c7*I   with c_n = (-1)^(n+1)/n ----
  const int eb = z2;            // free buffer
  const int w0i = yb;           // overwrite Y in place (own-element read->write)
  const int w1i = zb;
#pragma unroll
  for (int i = 0; i < 16; ++i) {
    const int e = tid + 256 * i;
    const int r = e >> 6, c = e & 63;
    const float ev = lds[yb + r * LDST + c] - ((r == c) ? 1.0f : 0.0f);
    lds[eb + r * LDST + c]  = ev;
    lds[w0i + r * LDST + c] = (-1.0f / 8.0f) * ev
                            + ((r == c) ? (1.0f / 7.0f) : 0.0f);
  }
  __syncthreads();

  // ---- Horner: W <- E*W + c_n I, n = 6..1 ----
  int w0 = w0i, w1 = w1i;
  for (int n = 0; n < 6; ++n) {
    const float cn = (n == 0) ? (-1.0f / 6.0f) : (n == 1) ? (1.0f / 5.0f)
                   : (n == 2) ? (-1.0f / 4.0f) : (n == 3) ? (1.0f / 3.0f)
                   : (n == 4) ? (-0.5f)        : 1.0f;
    gemm_tiles<2, EPI_DIAG>(lds, eb, w0, w1, nullptr,
                            wv & 3, (wv >> 2) * 2, lane, cn, 0.f);
    __syncthreads();
    const int t = w0; w0 = w1; w1 = t;
  }

  // ---- Out = 2^K * (E*W) + ln(s)*I, stored straight to global ----
  gemm_tiles<2, EPI_FINAL>(lds, eb, w0, 0, Ob,
                           wv & 3, (wv >> 2) * 2, lane,
                           (float)(1 << KSQRT), logs);
}

extern "C" void kernel_launch(void* const* d_in, const int* in_sizes, int n_in,
                              void* d_out, int out_size, void* d_ws, size_t ws_size,
                              hipStream_t stream) {
  (void)n_in; (void)out_size; (void)d_ws; (void)ws_size;
  const float* X = (const float*)d_in[0];
  float* Out = (float*)d_out;
  const int nb = in_sizes[0] / (MD * MD);   // 16384 matrices
  logm64_wmma_kernel<<<nb, 256, 0, stream>>>(X, Out);
}